// HBVMulTDET_36593121362026
// MI455X (gfx1250) — compile-verified
//
#include <hip/hip_runtime.h>
#include <cstdint>
#include <cstddef>

// ---------------------------------------------------------------------------
// HBV forward: 730-step scan over 10000 independent cells.
// Bandwidth-bound (~176 MB @ 23.3 TB/s => ~7.5us floor) with only ~313 waves
// of parallelism, so latency hiding comes from a 16-deep async global->LDS
// prefetch pipeline (CDNA5 GLOBAL_LOAD_ASYNC_TO_LDS_B32 / s_wait_asynccnt).
// ---------------------------------------------------------------------------

#define HBV_BLOCK  64
#define HBV_NSTAGE 16   // 16 stages * 3 async b32/lane-wave = 48 outstanding (<63 cap)

__device__ __forceinline__ void hbv_step(
    float Pt, float Tt, float PETt,
    float BETA, float FC, float K0, float K1, float K2, float LP,
    float PERCmax, float UZL, float TT, float CFMAX, float CFR, float CWH,
    float& SNOWPACK, float& MELTWATER, float& SM, float& SUZ, float& SLZ,
    float& Qsim, float& ETact)
{
    float is_rain = (Tt >= TT) ? 1.0f : 0.0f;
    float RAIN = Pt * is_rain;
    float SNOW = Pt - RAIN;
    SNOWPACK += SNOW;
    float melt = fminf(fmaxf(CFMAX * (Tt - TT), 0.0f), SNOWPACK);
    MELTWATER += melt;
    SNOWPACK  -= melt;
    float refreezing = fminf(fmaxf(CFR * CFMAX * (TT - Tt), 0.0f), MELTWATER);
    SNOWPACK  += refreezing;
    MELTWATER -= refreezing;
    float tosoil = fmaxf(MELTWATER - CWH * SNOWPACK, 0.0f);
    MELTWATER -= tosoil;
    // (SM/FC)^BETA with SM >= 1e-5 and FC >= 50 => argument strictly > 0.
    // v_log_f32 / v_exp_f32 (log2/exp2) hardware transcendentals.
    float sw = __builtin_amdgcn_exp2f(BETA * __builtin_amdgcn_logf(SM / FC));
    sw = fminf(fmaxf(sw, 0.0f), 1.0f);
    float recharge = (RAIN + tosoil) * sw;
    SM = SM + RAIN + tosoil - recharge;
    float excess = fmaxf(SM - FC, 0.0f);          // NOTE: reference does NOT remove excess from SM
    float evapfactor = fminf(fmaxf(SM / (LP * FC), 0.0f), 1.0f);
    float ETa = fminf(SM, PETt * evapfactor);
    SM = fmaxf(SM - ETa, 1e-5f);
    SUZ = SUZ + recharge + excess;
    float PERC = fminf(SUZ, PERCmax);
    SUZ -= PERC;
    float Q0 = K0 * fmaxf(SUZ - UZL, 0.0f);
    SUZ -= Q0;
    float Q1 = K1 * SUZ;
    SUZ -= Q1;
    SLZ += PERC;
    float Q2 = K2 * SLZ;
    SLZ -= Q2;
    Qsim  = Q0 + Q1 + Q2;
    ETact = ETa;
}

__global__ void __launch_bounds__(HBV_BLOCK)
hbv_forward_kernel(const float* __restrict__ x,       // (Nstep, Ngrid, 3)
                   const float* __restrict__ params,  // (Nstep, Ngrid, 12, 1)
                   float* __restrict__ out,           // [flow | AET | SWE], each Nstep*Ngrid
                   int Nstep, int Ngrid)
{
    __shared__ float xbuf[HBV_NSTAGE][HBV_BLOCK * 3];

    const int tid = threadIdx.x;
    const int g   = blockIdx.x * HBV_BLOCK + tid;
    if (g >= Ngrid) return;

    // Parameter bounds (BETA, FC, K0, K1, K2, LP, PERCmax, UZL, TT, CFMAX, CFR, CWH)
    const float lo[12] = {1.0f, 50.0f,   0.05f, 0.01f, 0.001f, 0.2f, 0.0f,  0.0f,   -2.5f, 0.5f,  0.0f, 0.0f};
    const float hi[12] = {6.0f, 1000.0f, 0.9f,  0.5f,  0.2f,   1.0f, 10.0f, 100.0f,  2.5f, 10.0f, 0.1f, 0.2f};

    // Only parameters[-1] (last timestep slice) is used by the reference.
    const float* pr = params + ((size_t)(Nstep - 1) * Ngrid + g) * 12;
    float ps[12];
    #pragma unroll
    for (int i = 0; i < 12; ++i) ps[i] = lo[i] + pr[i] * (hi[i] - lo[i]);
    const float BETA = ps[0],  FC      = ps[1],  K0  = ps[2],  K1   = ps[3];
    const float K2   = ps[4],  LP      = ps[5],  PERCmax = ps[6], UZL = ps[7];
    const float TT   = ps[8],  CFMAX   = ps[9],  CFR = ps[10], CWH  = ps[11];

    float SNOWPACK = 0.001f, MELTWATER = 0.001f, SM = 0.001f, SUZ = 0.001f, SLZ = 0.001f;

    const size_t SZ = (size_t)Nstep * (size_t)Ngrid;

    // Issue one pipeline stage: async-copy this lane's 12 bytes of x[t] into
    // its private LDS slot. Each lane consumes only data its own wave issued,
    // so ordering is enforced purely with s_wait_asynccnt (no barriers).
    auto issue_stage = [&](int t, int slot) {
        const float* gp = x + ((size_t)t * Ngrid + g) * 3;
        // Low 32 bits of a generic LDS pointer are the LDS byte offset.
        unsigned loff = (unsigned)(unsigned long long)(&xbuf[slot][tid * 3]);
        // INST_OFFSET applies to both global and LDS addresses.
        asm volatile("global_load_async_to_lds_b32 %0, %1, off"          :: "v"(loff), "v"(gp) : "memory");
        asm volatile("global_load_async_to_lds_b32 %0, %1, off offset:4" :: "v"(loff), "v"(gp) : "memory");
        asm volatile("global_load_async_to_lds_b32 %0, %1, off offset:8" :: "v"(loff), "v"(gp) : "memory");
    };

    // Prologue: fill the pipeline.
    const int npre = (HBV_NSTAGE < Nstep) ? HBV_NSTAGE : Nstep;
    for (int s = 0; s < npre; ++s) issue_stage(s, s);

    int t = 0;
    int steady_end = Nstep - HBV_NSTAGE;
    if (steady_end < 0) steady_end = 0;

    #pragma unroll 1
    for (; t < steady_end; ++t) {
        const int slot = t & (HBV_NSTAGE - 1);
        // Allow the newest (NSTAGE-1) stages (3 ops each) to stay in flight;
        // async loads complete in order, so stage t has landed in LDS.
        asm volatile("s_wait_asynccnt %0" :: "n"(3 * (HBV_NSTAGE - 1)) : "memory");
        float Pt   = xbuf[slot][tid * 3 + 0];
        float Tt   = xbuf[slot][tid * 3 + 1];
        float PETt = xbuf[slot][tid * 3 + 2];
        // Ensure the DS reads above retired before the refill overwrites the slot.
        asm volatile("s_wait_dscnt 0" ::: "memory");
        issue_stage(t + HBV_NSTAGE, slot);

        float Qsim, ETa;
        hbv_step(Pt, Tt, PETt, BETA, FC, K0, K1, K2, LP, PERCmax, UZL, TT, CFMAX, CFR, CWH,
                 SNOWPACK, MELTWATER, SM, SUZ, SLZ, Qsim, ETa);

        const size_t idx = (size_t)t * Ngrid + g;
        out[idx]          = Qsim;
        out[SZ + idx]     = ETa;
        out[2 * SZ + idx] = SNOWPACK;
    }

    // Drain: everything issued has a fixed-size tail; wait for all of it once.
    asm volatile("s_wait_asynccnt 0" ::: "memory");
    #pragma unroll 1
    for (; t < Nstep; ++t) {
        const int slot = t & (HBV_NSTAGE - 1);
        float Pt   = xbuf[slot][tid * 3 + 0];
        float Tt   = xbuf[slot][tid * 3 + 1];
        float PETt = xbuf[slot][tid * 3 + 2];

        float Qsim, ETa;
        hbv_step(Pt, Tt, PETt, BETA, FC, K0, K1, K2, LP, PERCmax, UZL, TT, CFMAX, CFR, CWH,
                 SNOWPACK, MELTWATER, SM, SUZ, SLZ, Qsim, ETa);

        const size_t idx = (size_t)t * Ngrid + g;
        out[idx]          = Qsim;
        out[SZ + idx]     = ETa;
        out[2 * SZ + idx] = SNOWPACK;
    }
}

extern "C" void kernel_launch(void* const* d_in, const int* in_sizes, int n_in,
                              void* d_out, int out_size, void* d_ws, size_t ws_size,
                              hipStream_t stream) {
    (void)n_in; (void)d_ws; (void)ws_size; (void)out_size;
    const float* x      = (const float*)d_in[0];
    const float* params = (const float*)d_in[1];
    float* out          = (float*)d_out;

    const int Nstep = 730;                               // reference: Nstep=730
    const int Ngrid = in_sizes[0] / (3 * Nstep);         // x is (Nstep, Ngrid, 3)

    dim3 block(HBV_BLOCK);
    dim3 grid((Ngrid + HBV_BLOCK - 1) / HBV_BLOCK);
    hbv_forward_kernel<<<grid, block, 0, stream>>>(x, params, out, Nstep, Ngrid);
}